// NTMInterface_69561290326017
// MI455X (gfx1250) — compile-verified
//
#include <hip/hip_runtime.h>
#include <hip/hip_bf16.h>
#include <math.h>

typedef float v2f __attribute__((ext_vector_type(2)));
typedef float v8f __attribute__((ext_vector_type(8)));

#define NB 128   // batch
#define NA 2048  // memory slots
#define NC 128   // slot width
#define NH 512   // hidden
#define NRD 2    // read heads
#define NPR 134  // params per read head
#define NPW 390  // write params
#define NJ 658   // 2*NPR + NPW total projection rows

__device__ __forceinline__ float sigmoidf_(float x) { return 1.0f / (1.0f + __expf(-x)); }
__device__ __forceinline__ float softplusf_(float x) { return (x > 20.0f) ? x : log1pf(__expf(x)); }

// Async DMA: global -> LDS, 16 bytes per active lane (gfx1250, tracked by ASYNCcnt).
// VDST operand is the LDS byte offset (flat->LDS mapping takes addr[31:0]).
__device__ __forceinline__ void async_g2l_b128(const void* ldsp, const float* g) {
  unsigned lds = (unsigned)(uintptr_t)ldsp;
  unsigned long long ga = (unsigned long long)(uintptr_t)g;
  asm volatile("global_load_async_to_lds_b128 %0, %1, off" :: "v"(lds), "v"(ga) : "memory");
}

// ---------------------------------------------------------------------------
// Kernel 1: projections  proj[j][b] = h[b,:] . Wall[j,:] + bias[j]
// One wave per 16x16 tile; V_WMMA_F32_16X16X4_F32, K looped over H=512.
// ---------------------------------------------------------------------------
__global__ __launch_bounds__(32) void ntm_proj_wmma(
    const float* __restrict__ h, const float* __restrict__ Wr, const float* __restrict__ br,
    const float* __restrict__ Ww, const float* __restrict__ bw, float* __restrict__ proj)
{
  const int b0   = blockIdx.x * 16;
  const int j0   = blockIdx.y * 16;
  const int lane = threadIdx.x & 31;
  const int m    = lane & 15;
  const int half = lane >> 4;
  const int j    = j0 + m;
  const bool valid = (j < NJ);

  const float* hrow = h + (size_t)(b0 + m) * NH + 2 * half;
  const float* wrow;
  if (j < NRD * NPR)  wrow = Wr + (size_t)j * NH + 2 * half;
  else if (valid)     wrow = Ww + (size_t)(j - NRD * NPR) * NH + 2 * half;
  else                wrow = Wr + 2 * half;                      // dummy (selected to 0)

  v8f d = {0.f, 0.f, 0.f, 0.f, 0.f, 0.f, 0.f, 0.f};
  for (int kk = 0; kk < NH; kk += 4) {
    v2f av; av.x = hrow[kk]; av.y = hrow[kk + 1];
    float t0 = wrow[kk], t1 = wrow[kk + 1];
    v2f bv; bv.x = valid ? t0 : 0.f; bv.y = valid ? t1 : 0.f;    // branch-free, EXEC all-ones
    d = __builtin_amdgcn_wmma_f32_16x16x4_f32(false, av, false, bv, (short)0, d, false, false);
  }
  if (valid) {
    const float bias = (j < NRD * NPR) ? br[j] : bw[j - NRD * NPR];
    const int mb = b0 + half * 8;
#pragma unroll
    for (int i = 0; i < 8; ++i)
      proj[(size_t)j * NB + mb + i] = d[i] + bias;
  }
}

// ---------------------------------------------------------------------------
// Kernel 2: activations per (batch, head).
// ---------------------------------------------------------------------------
__global__ __launch_bounds__(128) void ntm_act(
    const float* __restrict__ proj, float* __restrict__ Q,
    float* __restrict__ scal, float* __restrict__ EA)
{
  __shared__ float red[128];
  const int b = blockIdx.x;
  const int c = threadIdx.x;
  for (int k = 0; k < 3; ++k) {
    const int base = (k < 2) ? k * NPR : NRD * NPR;
    const float qa = sigmoidf_(proj[(size_t)(base + c) * NB + b]);
    Q[((size_t)b * 3 + k) * NC + c] = qa;
    red[c] = qa * qa; __syncthreads();
    for (int off = 64; off > 0; off >>= 1) {
      if (c < off) red[c] += red[c + off];
      __syncthreads();
    }
    if (c == 0) {
      const float qn    = fmaxf(sqrtf(red[0]), 1e-8f);
      const float beta  = softplusf_(proj[(size_t)(base + NC    ) * NB + b]) + 1.f;
      const float gate  = sigmoidf_ (proj[(size_t)(base + NC + 1) * NB + b]);
      const float s0 = proj[(size_t)(base + NC + 2) * NB + b];
      const float s1 = proj[(size_t)(base + NC + 3) * NB + b];
      const float s2 = proj[(size_t)(base + NC + 4) * NB + b];
      const float gamma = softplusf_(proj[(size_t)(base + NC + 5) * NB + b]) + 1.f;
      const float mx = fmaxf(s0, fmaxf(s1, s2));
      const float e0 = __expf(s0 - mx), e1 = __expf(s1 - mx), e2 = __expf(s2 - mx);
      const float inv = 1.f / (e0 + e1 + e2);
      float* scp = scal + ((size_t)b * 3 + k) * 8;
      scp[0] = qn; scp[1] = beta; scp[2] = gate; scp[3] = gamma;
      scp[4] = e0 * inv; scp[5] = e1 * inv; scp[6] = e2 * inv; scp[7] = 0.f;
    }
    __syncthreads();
  }
  EA[((size_t)b * 2    ) * NC + c] = sigmoidf_(proj[(size_t)(NRD * NPR + 134 + c) * NB + b]);
  EA[((size_t)b * 2 + 1) * NC + c] = sigmoidf_(proj[(size_t)(NRD * NPR + 262 + c) * NB + b]);
}

__device__ __forceinline__ float blockSum256(float v, float* red) {
  const int tid = threadIdx.x;
  red[tid] = v; __syncthreads();
#pragma unroll
  for (int off = 128; off > 0; off >>= 1) {
    if (tid < off) red[tid] += red[tid + off];
    __syncthreads();
  }
  const float r = red[0]; __syncthreads();
  return r;
}
__device__ __forceinline__ float blockMax256(float v, float* red) {
  const int tid = threadIdx.x;
  red[tid] = v; __syncthreads();
#pragma unroll
  for (int off = 128; off > 0; off >>= 1) {
    if (tid < off) red[tid] = fmaxf(red[tid], red[tid + off]);
    __syncthreads();
  }
  const float r = red[0]; __syncthreads();
  return r;
}

// ---------------------------------------------------------------------------
// Kernel 3: one workgroup (256 thr = 8 waves) per batch b.
//  - queries staged once into LDS via async DMA (B fragments then come from DS)
//  - prev attention double-buffered via GLOBAL_LOAD_ASYNC_TO_LDS_B128, with the
//    DMA for head k+1 overlapping head k's softmax pipeline (ASYNCcnt in-order:
//    s_wait_asynccnt 2 retires the current head's 2 wave-instructions)
//  - Pass A: WMMA similarity + piggybacked row norms
//  - Pass B: fused read-vector contraction + erase/add update, coalesced
// ---------------------------------------------------------------------------
__global__ __launch_bounds__(256) void ntm_addr_update(
    const float* __restrict__ mem, const float* __restrict__ ra, const float* __restrict__ wa,
    const float* __restrict__ Q, const float* __restrict__ scal, const float* __restrict__ EA,
    float* __restrict__ out)
{
  __shared__ float s_w[3][NA];      // sim -> final attention weights (24 KB)
  __shared__ float s_mn[NA];        // memory row norms (8 KB)
  __shared__ float s_tmp[NA];       // gated weights for shift conv (8 KB)
  __shared__ float s_red[256];      // reductions (1 KB)
  __shared__ float s_q[3 * NC];     // staged queries (1.5 KB)
  __shared__ float s_prev[2][NA];   // double-buffered prev attention (16 KB)

  const int b    = blockIdx.x;
  const int tid  = threadIdx.x;
  const int lane = tid & 31;
  const int wave = tid >> 5;
  const float* memb = mem + (size_t)b * NA * NC;
  const float* Qb   = Q + (size_t)b * 3 * NC;

  // ---- stage queries (async DMA), needed by Pass A ----
  if (tid < (3 * NC) / 4)
    async_g2l_b128(&s_q[4 * tid], Qb + 4 * tid);
  asm volatile("s_wait_asynccnt 0" ::: "memory");
  __syncthreads();

  // ---- kick off prev[head 0] DMA; overlaps with the whole WMMA pass ----
  {
    const float* p0 = ra + (size_t)b * NA;   // head 0 = read head r=0
    async_g2l_b128(&s_prev[0][4 * tid],         p0 + 4 * tid);
    async_g2l_b128(&s_prev[0][4 * (tid + 256)], p0 + 4 * (tid + 256));
  }

  // ---- Pass A: sim via WMMA + row norms ----
  {
    const int mrow = lane & 15;
    const int half = lane >> 4;
    const int n    = lane & 15;                  // WMMA N index = head (0..2 valid)
    const bool nval = (n < 3);
    const float* qrow = nval ? (s_q + n * NC + 2 * half) : (s_q + 2 * half);
    for (int tile = wave; tile < NA / 16; tile += 8) {
      const int a0 = tile * 16;
      const float* arow = memb + (size_t)(a0 + mrow) * NC + 2 * half;
      v8f d = {0.f, 0.f, 0.f, 0.f, 0.f, 0.f, 0.f, 0.f};
      float ss = 0.f;
      for (int kk = 0; kk < NC; kk += 4) {
        v2f av; av.x = arow[kk]; av.y = arow[kk + 1];
        ss += av.x * av.x + av.y * av.y;          // row norm piggybacks on A loads
        float t0 = qrow[kk], t1 = qrow[kk + 1];   // LDS (ds_load) B fragments
        v2f bv; bv.x = nval ? t0 : 0.f; bv.y = nval ? t1 : 0.f;
        d = __builtin_amdgcn_wmma_f32_16x16x4_f32(false, av, false, bv, (short)0, d, false, false);
      }
      const float tot = ss + __shfl_xor(ss, 16, 32);
      if (lane < 16) s_mn[a0 + mrow] = fmaxf(sqrtf(tot), 1e-8f);
      if (n < 3) {
        const int mb = half * 8;
#pragma unroll
        for (int i = 0; i < 8; ++i) s_w[n][a0 + mb + i] = d[i];
      }
    }
  }
  __syncthreads();

  // ---- Addressing per head, with double-buffered async prev ----
  for (int k = 0; k < 3; ++k) {
    if (k < 2) {  // prefetch next head's prev attention into the other buffer
      const float* pn = (k + 1 < 2) ? (ra + ((size_t)(k + 1) * NB + b) * NA)
                                    : (wa + (size_t)b * NA);
      float* dst = s_prev[(k + 1) & 1];
      async_g2l_b128(&dst[4 * tid],         pn + 4 * tid);
      async_g2l_b128(&dst[4 * (tid + 256)], pn + 4 * (tid + 256));
      asm volatile("s_wait_asynccnt 2" ::: "memory");  // current head's buffer landed
    } else {
      asm volatile("s_wait_asynccnt 0" ::: "memory");
    }
    __syncthreads();
    const float* prev = s_prev[k & 1];

    const float* sc = scal + ((size_t)b * 3 + k) * 8;
    const float qn = sc[0], beta = sc[1], gate = sc[2], gamma = sc[3];
    const float sh0 = sc[4], sh1 = sc[5], sh2 = sc[6];

    float xs[8];
    float lmax = -3.4e38f;
#pragma unroll
    for (int i = 0; i < 8; ++i) {
      const int a = tid + i * 256;
      xs[i] = beta * (s_w[k][a] / (qn * s_mn[a]));
      lmax = fmaxf(lmax, xs[i]);
    }
    const float gmax = blockMax256(lmax, s_red);
    float lsum = 0.f;
#pragma unroll
    for (int i = 0; i < 8; ++i) { xs[i] = __expf(xs[i] - gmax); lsum += xs[i]; }
    const float inv = 1.f / blockSum256(lsum, s_red);
#pragma unroll
    for (int i = 0; i < 8; ++i) {
      const int a = tid + i * 256;
      s_tmp[a] = gate * xs[i] * inv + (1.f - gate) * prev[a];   // gated interpolation
    }
    __syncthreads();
    float psum = 0.f;
#pragma unroll
    for (int i = 0; i < 8; ++i) {
      const int a = tid + i * 256;   // circular shift conv, offsets +1,0,-1
      const float wsv = s_tmp[(a + 1) & (NA - 1)] * sh0 + s_tmp[a] * sh1 + s_tmp[(a - 1) & (NA - 1)] * sh2;
      xs[i] = powf(wsv, gamma);
      psum += xs[i];
    }
    const float invp = 1.f / (blockSum256(psum, s_red) + 1e-12f);
#pragma unroll
    for (int i = 0; i < 8; ++i) {
      const int a = tid + i * 256;
      s_w[k][a] = xs[i] * invp;     // final attention weights
    }
    __syncthreads();
  }

  // ---- Pass B: fused read vectors + memory update ----
  {
    const int c     = tid & (NC - 1);
    const int halfb = tid >> 7;
    const float* EAb = EA + (size_t)b * 2 * NC;
    const float ec = EAb[c];
    const float ac = EAb[NC + c];
    float* nm = out + (size_t)NRD * NB * NC + (size_t)b * NA * NC;
    float rv0 = 0.f, rv1 = 0.f;
    for (int a = halfb; a < NA; a += 2) {
      if (a + 8 < NA) __builtin_prefetch(memb + (size_t)(a + 8) * NC + c, 0, 1);
      const float mv = memb[(size_t)a * NC + c];
      const float ww = s_w[2][a];
      rv0 += s_w[0][a] * mv;
      rv1 += s_w[1][a] * mv;
      nm[(size_t)a * NC + c] = mv * (1.f - ww * ec) + ww * ac;
    }
    s_tmp[tid]       = rv0;
    s_tmp[256 + tid] = rv1;
    __syncthreads();
    if (tid < NC) {
      out[(size_t)0 * NB * NC + (size_t)b * NC + tid] = s_tmp[tid] + s_tmp[tid + 128];
      out[(size_t)1 * NB * NC + (size_t)b * NC + tid] = s_tmp[256 + tid] + s_tmp[256 + tid + 128];
    }
  }
}

extern "C" void kernel_launch(void* const* d_in, const int* in_sizes, int n_in,
                              void* d_out, int out_size, void* d_ws, size_t ws_size,
                              hipStream_t stream) {
  const float* h   = (const float*)d_in[0];
  const float* mem = (const float*)d_in[1];
  const float* ra  = (const float*)d_in[2];
  const float* wa  = (const float*)d_in[3];
  const float* Wr  = (const float*)d_in[4];
  const float* br  = (const float*)d_in[5];
  const float* Ww  = (const float*)d_in[6];
  const float* bw  = (const float*)d_in[7];
  float* out = (float*)d_out;

  float* ws   = (float*)d_ws;
  float* proj = ws;                          // NJ * NB
  float* Q    = proj + (size_t)NJ * NB;      // NB * 3 * NC
  float* scal = Q    + (size_t)NB * 3 * NC;  // NB * 3 * 8
  float* EA   = scal + (size_t)NB * 3 * 8;   // NB * 2 * NC   (~660 KB total)

  ntm_proj_wmma<<<dim3(NB / 16, (NJ + 15) / 16), 32, 0, stream>>>(h, Wr, br, Ww, bw, proj);
  ntm_act<<<NB, 128, 0, stream>>>(proj, Q, scal, EA);
  ntm_addr_update<<<NB, 256, 0, stream>>>(mem, ra, wa, Q, scal, EA, out);
}